// AnchorTarget_40149354283769
// MI455X (gfx1250) — compile-verified
//
#include <hip/hip_runtime.h>
#include <cstdint>

// Match JAX fp32 numerics: no FMA contraction anywhere (labels/argmax are
// threshold/tie sensitive).
#pragma clang fp contract(off)

#define NA 147456   // 128*128*9 anchors
#define NG 256      // gt boxes
#define NCELL 16384 // 128*128

// Classic py-faster-rcnn base anchors for base=16, ratios(.5,1,2), scales(8,16,32)
// (exact integers; verified against the reference's banker's-rounding math).
static constexpr float kBX1[9] = {-84.f,-176.f,-360.f,-56.f,-120.f,-248.f,-36.f,-80.f,-168.f};
static constexpr float kBY1[9] = {-40.f,-88.f,-184.f,-56.f,-120.f,-248.f,-80.f,-168.f,-344.f};
static constexpr float kBX2[9] = { 99.f, 191.f, 375.f, 71.f, 135.f, 263.f, 51.f, 95.f, 183.f};
static constexpr float kBY2[9] = { 55.f, 103.f, 199.f, 71.f, 135.f, 263.f, 95.f, 183.f, 359.f};

__device__ __forceinline__ void anchor_coords(int a, float& x1, float& y1,
                                              float& x2, float& y2) {
  unsigned ua = (unsigned)a;
  unsigned cell = ua / 9u;
  unsigned k = ua - cell * 9u;
  float sx = (float)((cell & 127u) * 16u);   // x shift from column
  float sy = (float)((cell >> 7)   * 16u);   // y shift from row
  x1 = kBX1[k] + sx; y1 = kBY1[k] + sy;
  x2 = kBX2[k] + sx; y2 = kBY2[k] + sy;
}

// -------- Kernel A: per-anchor max/argmax IoU over gts + bbox targets --------
__global__ void __launch_bounds__(256)
anchor_pass(const float* __restrict__ gt, const float* __restrict__ meta,
            float* __restrict__ out_bbox, float* __restrict__ maxov) {
  __shared__ float4 sg[NG];
  const int tid = threadIdx.x;

  // CDNA5 async DMA: stage the 256 gt boxes (4KB) global -> LDS, bypassing VGPRs.
  {
    uint32_t lds_off = (uint32_t)(uintptr_t)(&sg[tid]);      // low 32b = LDS offset
    const float4* src = (const float4*)gt + tid;
    asm volatile("global_load_async_to_lds_b128 %0, %1, off"
                 :: "v"(lds_off), "v"(src) : "memory");
    asm volatile("s_wait_asynccnt 0x0" ::: "memory");
  }
  __syncthreads();

  const int a = blockIdx.x * 256 + tid;
  float x1, y1, x2, y2;
  anchor_coords(a, x1, y1, x2, y2);
  const float area_a = (x2 - x1 + 1.0f) * (y2 - y1 + 1.0f);

  float best = -1.0f;
  int bi = 0;
  for (int g = 0; g < NG; ++g) {
    float4 gb = sg[g];
    float area_b = (gb.z - gb.x + 1.0f) * (gb.w - gb.y + 1.0f);
    float iw = fminf(x2, gb.z) - fmaxf(x1, gb.x) + 1.0f;
    float ih = fminf(y2, gb.w) - fmaxf(y1, gb.y) + 1.0f;
    iw = fmaxf(iw, 0.0f);
    ih = fmaxf(ih, 0.0f);
    float inter = iw * ih;
    float uni = fmaxf(area_a + area_b - inter, 1e-8f);
    float v = inter / uni;
    if (v > best) { best = v; bi = g; }   // strict > == jnp.argmax first-occurrence
  }
  maxov[a] = best;

  const float H = meta[0], W = meta[1];
  bool inside = (x1 >= 0.0f) && (y1 >= 0.0f) && (x2 < W) && (y2 < H);

  float4 gb = sg[bi];
  float ew = x2 - x1 + 1.0f, eh = y2 - y1 + 1.0f;
  float ecx = x1 + 0.5f * ew, ecy = y1 + 0.5f * eh;
  float gw = gb.z - gb.x + 1.0f, gh = gb.w - gb.y + 1.0f;
  float gcx = gb.x + 0.5f * gw, gcy = gb.y + 0.5f * gh;
  float4 bb;
  bb.x = (gcx - ecx) / ew;
  bb.y = (gcy - ecy) / eh;
  bb.z = logf(gw / ew);
  bb.w = logf(gh / eh);
  if (!inside) { bb.x = 0.0f; bb.y = 0.0f; bb.z = 0.0f; bb.w = 0.0f; }
  ((float4*)out_bbox)[a] = bb;
}

// -------- Kernel B: per-gt argmax over (inside ? iou : -1) across anchors ----
__global__ void __launch_bounds__(256)
gt_argmax_pass(const float* __restrict__ gt, const float* __restrict__ meta,
               int* __restrict__ gtamax) {
  const int g = blockIdx.x;
  const int tid = threadIdx.x;
  const float H = meta[0], W = meta[1];
  const float gx1 = gt[g * 4 + 0], gy1 = gt[g * 4 + 1];
  const float gx2 = gt[g * 4 + 2], gy2 = gt[g * 4 + 3];
  const float area_b = (gx2 - gx1 + 1.0f) * (gy2 - gy1 + 1.0f);

  float best = -2.0f;
  int bi = 0;
  for (int cell = tid; cell < NCELL; cell += 256) {
    float sx = (float)((cell & 127) * 16);
    float sy = (float)((cell >> 7) * 16);
#pragma unroll
    for (int k = 0; k < 9; ++k) {        // base anchors fold to immediates
      float x1 = kBX1[k] + sx, y1 = kBY1[k] + sy;
      float x2 = kBX2[k] + sx, y2 = kBY2[k] + sy;
      bool inside = (x1 >= 0.0f) && (y1 >= 0.0f) && (x2 < W) && (y2 < H);
      float area_a = (x2 - x1 + 1.0f) * (y2 - y1 + 1.0f);
      float iw = fminf(x2, gx2) - fmaxf(x1, gx1) + 1.0f;
      float ih = fminf(y2, gy2) - fmaxf(y1, gy1) + 1.0f;
      iw = fmaxf(iw, 0.0f);
      ih = fmaxf(ih, 0.0f);
      float inter = iw * ih;
      float uni = fmaxf(area_a + area_b - inter, 1e-8f);
      float v = inter / uni;
      v = inside ? v : -1.0f;
      int a = cell * 9 + k;              // strictly increasing per-thread scan
      if (v > best) { best = v; bi = a; }
    }
  }

  __shared__ float sv[256];
  __shared__ int   si[256];
  sv[tid] = best; si[tid] = bi;
  __syncthreads();
  for (int s = 128; s > 0; s >>= 1) {
    if (tid < s) {
      float v2 = sv[tid + s]; int i2 = si[tid + s];
      if (v2 > sv[tid] || (v2 == sv[tid] && i2 < si[tid])) {  // tie -> lowest index
        sv[tid] = v2; si[tid] = i2;
      }
    }
    __syncthreads();
  }
  if (tid == 0) gtamax[g] = si[0];
}

// -------- Kernel C: finalize labels (exact .at[].add duplicate semantics) ----
__global__ void __launch_bounds__(256)
labels_pass(const float* __restrict__ meta, const float* __restrict__ maxov,
            const int* __restrict__ gtamax, float* __restrict__ out_labels) {
  __shared__ int sga[NG];
  const int tid = threadIdx.x;
  sga[tid] = gtamax[tid];
  __syncthreads();

  const int a = blockIdx.x * 256 + tid;
  const float mo = maxov[a];

  int c = 0;
#pragma unroll 8
  for (int j = 0; j < NG; ++j) c += (sga[j] == a) ? 1 : 0;

  float v = (mo < 0.3f) ? 0.0f : -1.0f;          // pre-scatter labels
  float lab = v + (float)c * (1.0f - v);         // scatter-add with duplicates
  if (mo >= 0.7f) lab = 1.0f;                    // positive override (after scatter)

  float x1, y1, x2, y2;
  anchor_coords(a, x1, y1, x2, y2);
  const float H = meta[0], W = meta[1];
  bool inside = (x1 >= 0.0f) && (y1 >= 0.0f) && (x2 < W) && (y2 < H);
  if (!inside) lab = -1.0f;                      // final outside mask

  out_labels[a] = lab;
}

extern "C" void kernel_launch(void* const* d_in, const int* in_sizes, int n_in,
                              void* d_out, int out_size, void* d_ws, size_t ws_size,
                              hipStream_t stream) {
  // inputs: d_in[0]=scores (unused, shape only), d_in[1]=gt_boxes (1,256,4),
  //         d_in[2]=metadata (1,3) = {H, W, scale}
  const float* gt   = (const float*)d_in[1];
  const float* meta = (const float*)d_in[2];

  float* out_labels = (float*)d_out;          // (1, 147456)
  float* out_bbox   = (float*)d_out + NA;     // (1, 147456, 4)

  float* maxov  = (float*)d_ws;                               // NA floats
  int*   gtamax = (int*)((char*)d_ws + (size_t)NA * sizeof(float)); // 256 ints

  anchor_pass   <<<NA / 256, 256, 0, stream>>>(gt, meta, out_bbox, maxov);
  gt_argmax_pass<<<NG,       256, 0, stream>>>(gt, meta, gtamax);
  labels_pass   <<<NA / 256, 256, 0, stream>>>(meta, maxov, gtamax, out_labels);
}